// FFM_58574763983366
// MI455X (gfx1250) — compile-verified
//
#include <hip/hip_runtime.h>
#include <math.h>

typedef __attribute__((ext_vector_type(2))) float v2f;
typedef __attribute__((ext_vector_type(8))) float v8f;

#define N      512     // features
#define NF     30      // fields
#define NK     40      // factor dim
#define B_ROWS 32768   // batch

// XOR bank swizzle: keeps float4/float2 blocks contiguous (XOR value is a
// multiple of 4) while making rows land in distinct banks for the column-
// parallel access patterns below.
__device__ __forceinline__ int swz(int r, int c) {
  return r * N + (c ^ ((r & 15) << 2));
}

// ---------------------------------------------------------------------------
// Kernel 1: build M = 0.5*C (zero diag), stored k-pair interleaved:
//   Mp[(k>>1)*N + j] = float2( M[k][j], M[k+1][j] )
// so the main kernel's B-fragment fetch is one global_load_b64 per lane.
// C[i,j] = sum_k v[i, field(j), k] * v[j, field(i), k]  (symmetric)
// ---------------------------------------------------------------------------
__global__ __launch_bounds__(256)
void ffm_prep_kernel(const float* __restrict__ v, const int* __restrict__ f2f,
                     float* __restrict__ Mp) {
  const int idx = blockIdx.x * 256 + threadIdx.x;   // grid sized exactly N*N
  const int i = idx >> 9;           // k-row of M
  const int j = idx & (N - 1);      // column
  float s = 0.0f;
  if (i != j) {
    const int fi = f2f[i];
    const int fj = f2f[j];
    const float* vi = v + ((size_t)i * NF + fj) * NK;
    const float* vj = v + ((size_t)j * NF + fi) * NK;
#pragma unroll
    for (int kk = 0; kk < NK; ++kk) s = fmaf(vi[kk], vj[kk], s);
    s *= 0.5f;
  }
  Mp[((size_t)(i >> 1) * N + j) * 2 + (i & 1)] = s;
}

// ---------------------------------------------------------------------------
// Kernel 2: out[b] = sigmoid( b0 + X[b]·w1 + X[b]^T M X[b] )
// One wave32 per 32 rows (two 16-row tiles sharing each B fragment).
// Quadratic form via V_WMMA_F32_16X16X4_F32 (fp32-exact vs reference).
// ---------------------------------------------------------------------------
__global__ __launch_bounds__(32)
void ffm_main_kernel(const float* __restrict__ X, const float* __restrict__ w1,
                     const float* __restrict__ bias, const float* __restrict__ Mp,
                     float* __restrict__ out) {
  __shared__ float Xs[32 * N];      // exactly 64 KB, XOR-swizzled rows

  const int lane    = threadIdx.x;  // 0..31
  const int rowBase = blockIdx.x * 32;

  // ---- Stage X stripe (32 x 512) into LDS (float4, swizzled) ----
  {
    const float4* src = (const float4*)(X + (size_t)rowBase * N);
    for (int t = lane; t < 32 * (N / 4); t += 32) {
      const int r  = t >> 7;          // 128 float4 per row
      const int c4 = t & 127;
      float4 val = src[t];            // linear index r*128 + c4 == t
      *(float4*)&Xs[swz(r, c4 * 4)] = val;
    }
  }
  __syncthreads();

  // ---- Linear term: lane l computes dot(X[row l], w1), float4 at a time ----
  float lin = 0.0f;
  {
    const float* Xrow = Xs + lane * N;
    const int sL = (lane & 15) << 2;
    for (int i = 0; i < N; i += 4) {
      float4 xv = *(const float4*)(Xrow + (i ^ sL));   // swizzle keeps 4-blocks
      float4 wv = *(const float4*)(w1 + i);
      lin = fmaf(xv.x, wv.x,
            fmaf(xv.y, wv.y,
            fmaf(xv.z, wv.z,
            fmaf(xv.w, wv.w, lin))));
    }
  }

  // ---- Quadratic term ----
  // A fragment (16x4 f32): lanes 0-15 -> K = k0+0/1 ; lanes 16-31 -> K = k0+2/3
  const int g    = lane >> 4;       // half-wave id
  const int cOff = g << 1;          // 0 or 2
  const int jl   = lane & 15;
  const int rOff = g << 3;          // C/D layout: VGPR m <-> row m (+8 hi half)
  const int sA   = jl << 2;         // column swizzle for rows with (r&15)==jl

  float rs0[8], rs1[8];
#pragma unroll
  for (int m = 0; m < 8; ++m) { rs0[m] = 0.0f; rs1[m] = 0.0f; }

  const v2f*   M2    = (const v2f*)Mp;
  const float* XrowA = Xs + jl * N;   // tile0 A-row; tile1 is +16*N (32768 B: fits DS imm)

  for (int jt = 0; jt < N / 16; ++jt) {
    const int jcol = jt * 16 + jl;
    v8f acc0 = {};
    v8f acc1 = {};
#pragma unroll 4
    for (int k0 = 0; k0 < N; k0 += 4) {
      const int kc = (k0 + cOff) ^ sA;                 // pair-aligned (even)
      v2f a0 = *(const v2f*)(XrowA + kc);              // ds_load_b64, even pair
      v2f a1 = *(const v2f*)(XrowA + 16 * N + kc);     // same addr + 32768B
      v2f b  = M2[(size_t)((k0 >> 1) + g) * N + jcol]; // one global b64, used 2x
      acc0 = __builtin_amdgcn_wmma_f32_16x16x4_f32(false, a0, false, b,
                                                   (short)0, acc0, false, false);
      acc1 = __builtin_amdgcn_wmma_f32_16x16x4_f32(false, a1, false, b,
                                                   (short)0, acc1, false, false);
    }
    // fold Y[r, jcol] * X[r, jcol] into per-row partial sums
#pragma unroll
    for (int m = 0; m < 8; ++m) {
      const int cs = jcol ^ ((m + rOff) << 2);
      rs0[m] = fmaf(acc0[m], Xs[(m + rOff) * N + cs],        rs0[m]);
      rs1[m] = fmaf(acc1[m], Xs[(16 + m + rOff) * N + cs],   rs1[m]);
    }
  }

  // ---- Reduce over the 16 lanes (columns) sharing each row set ----
#pragma unroll
  for (int m = 0; m < 8; ++m) {
    float s0 = rs0[m];
    float s1 = rs1[m];
    s0 += __shfl_xor(s0, 1, 16);
    s0 += __shfl_xor(s0, 2, 16);
    s0 += __shfl_xor(s0, 4, 16);
    s0 += __shfl_xor(s0, 8, 16);
    s1 += __shfl_xor(s1, 1, 16);
    s1 += __shfl_xor(s1, 2, 16);
    s1 += __shfl_xor(s1, 4, 16);
    s1 += __shfl_xor(s1, 8, 16);
    rs0[m] = s0;
    rs1[m] = s1;
  }

  // ---- Hand 8-row groups to the right lanes via LDS (reuse Xs) ----
  float* red = Xs;                  // X data no longer needed
  __syncthreads();
  if (lane == 0) {
#pragma unroll
    for (int m = 0; m < 8; ++m) { red[m]      = rs0[m]; red[16 + m] = rs1[m]; }
  }
  if (lane == 16) {
#pragma unroll
    for (int m = 0; m < 8; ++m) { red[8 + m]  = rs0[m]; red[24 + m] = rs1[m]; }
  }
  __syncthreads();

  const float logit = lin + red[lane] + bias[0];
  out[rowBase + lane] = 1.0f / (1.0f + expf(-logit));
}

// ---------------------------------------------------------------------------
extern "C" void kernel_launch(void* const* d_in, const int* in_sizes, int n_in,
                              void* d_out, int out_size, void* d_ws, size_t ws_size,
                              hipStream_t stream) {
  const float* X   = (const float*)d_in[0];   // (B, n) f32
  const float* w1  = (const float*)d_in[1];   // (n, 1) f32
  const float* b   = (const float*)d_in[2];   // (1,)  f32
  const float* v   = (const float*)d_in[3];   // (n, f, k) f32
  const int*   f2f = (const int*)d_in[4];     // (n,) i32
  float*       out = (float*)d_out;           // (B,) f32
  float*       Mp  = (float*)d_ws;            // n*n f32 = 1 MB scratch (k-pair interleaved)

  ffm_prep_kernel<<<(N * N) / 256, 256, 0, stream>>>(v, f2f, Mp);
  ffm_main_kernel<<<B_ROWS / 32, 32, 0, stream>>>(X, w1, b, Mp, out);
}